// SSKernelNPLR_730144440569
// MI455X (gfx1250) — compile-verified
//
#include <hip/hip_runtime.h>
#include <hip/hip_bf16.h>

// S4 NPLR kernel generation: Cauchy kernel + Woodbury + irfft (f32 WMMA GEMM).
// Shapes: CH=1, H=S=256, N=64, R=1, L=2048, Lh=1025.
// Workspace layout:
//   A      : 256 x 2052 f32  (interleaved Re/Im spectrum, K-padded)  = 2,101,248 B
//   Bbasis : 2052 x 2048 f32 (weighted cos/-sin irfft basis)         = 16,809,984 B

#define HH   256
#define NN   64
#define LL   2048
#define LH   1025      // L/2 + 1
#define KP   2052      // 2*LH padded to multiple of 4
#define TWO_PI 6.283185307179586f

typedef __attribute__((ext_vector_type(2))) float v2f;
typedef __attribute__((ext_vector_type(8))) float v8f;
typedef __attribute__((ext_vector_type(4))) unsigned int u32x4;
typedef __attribute__((ext_vector_type(4))) int i32x4;
typedef __attribute__((ext_vector_type(8))) int i32x8;

// ---------------------------------------------------------------------------
// Stage 0: weighted irfft basis.
// row kk = 2*l   :  w_l * cos(2*pi*l*t/L)
// row kk = 2*l+1 : -w_l * sin(2*pi*l*t/L)
// w_l = 1/L for l==0 or l==L/2, else 2/L (matches np.fft.irfft Hermitian sum).
// Phase reduced mod L in ints so the float angle stays small/exact.
// ---------------------------------------------------------------------------
__global__ void ssk_build_basis(float* __restrict__ Bb) {
    int idx = blockIdx.x * blockDim.x + threadIdx.x;
    if (idx >= KP * LL) return;
    int t  = idx & (LL - 1);
    int kk = idx >> 11;              // LL == 2048 == 2^11
    int l  = kk >> 1;
    float val = 0.0f;
    if (l <= LL / 2) {
        float w = (l == 0 || l == LL / 2) ? (1.0f / (float)LL) : (2.0f / (float)LL);
        int phase = (l * t) & (LL - 1);
        float ang = (TWO_PI / (float)LL) * (float)phase;
        val = (kk & 1) ? (-w * sinf(ang)) : (w * cosf(ang));
    }
    Bb[idx] = val;
}

// ---------------------------------------------------------------------------
// Stage 1: Cauchy sum over 2N conjugate-extended poles + Woodbury (R=1)
// + bilinear 2/(1+omega) factor. blockIdx.x = head h, blockIdx.y = 256-freq chunk; one thread per freq l.
// Per-head pole data staged in LDS: wdt and v_ab = Bcat[a]*Ccat[b] (4 pairs).
// ---------------------------------------------------------------------------
__global__ void ssk_cauchy(const float* __restrict__ Cm,  // (1,H,N,2)
                           const float* __restrict__ Bm,  // (1,S,N,2)
                           const float* __restrict__ Pm,  // (R,S,N,2)
                           const float* __restrict__ iwr, // (S,N)
                           const float* __restrict__ wim, // (S,N)
                           const float* __restrict__ ldt, // (H,)
                           float* __restrict__ A) {       // (H, KP)
    __shared__ float sh[10][NN];
    const int h   = blockIdx.x;
    const int tid = threadIdx.x;
    const float dt = __expf(ldt[h]);

    if (tid < NN) {
        const int n = tid;
        const int base = h * NN + n;        // S == H so s == h
        const float wr = -__expf(iwr[base]) * dt;   // Re(w)*dt
        const float wi = wim[base] * dt;            // Im(w)*dt
        const float br = Bm[2 * base], bi = Bm[2 * base + 1];
        const float pr = Pm[2 * base], pj = Pm[2 * base + 1];
        const float cr = Cm[2 * base], ci = Cm[2 * base + 1];
        const float qr = pr, qi = -pj;              // Q = conj(P)
        sh[0][n] = wr;  sh[1][n] = wi;
        sh[2][n] = br * cr - bi * ci;  sh[3][n] = br * ci + bi * cr;   // v00 = B*C
        sh[4][n] = br * qr - bi * qi;  sh[5][n] = br * qi + bi * qr;   // v01 = B*Q
        sh[6][n] = pr * cr - pj * ci;  sh[7][n] = pr * ci + pj * cr;   // v10 = P*C
        sh[8][n] = pr * qr - pj * qi;  sh[9][n] = pr * qi + pj * qr;   // v11 = P*Q
    }
    __syncthreads();

    const int l = blockIdx.y * blockDim.x + tid;
    if (l > LH) return;
    if (l == LH) {            // zero the K-padding columns of A
        A[h * KP + 2 * LH]     = 0.0f;
        A[h * KP + 2 * LH + 1] = 0.0f;
        return;
    }

    // omega = exp(-2*pi*i*l/L); z = 2*(1-omega)/(1+omega)
    float s, c;
    sincosf(-(TWO_PI / (float)LL) * (float)l, &s, &c);
    const float dr1 = 1.0f + c, di1 = s;                 // 1 + omega
    const float invm = 1.0f / (dr1 * dr1 + di1 * di1);
    const float nr = 2.0f - 2.0f * c, ni = -2.0f * s;    // 2*(1 - omega)
    const float zr = (nr * dr1 + ni * di1) * invm;
    const float zi = (ni * dr1 - nr * di1) * invm;

    float r00r = 0, r00i = 0, r01r = 0, r01i = 0;
    float r10r = 0, r10i = 0, r11r = 0, r11i = 0;

    #pragma unroll 4
    for (int n = 0; n < NN; ++n) {
        const float wr = sh[0][n], wi = sh[1][n];
        const float dr  = zr - wr;          // both denominators share the real part
        const float diA = zi - wi;          // z - w
        const float diB = zi + wi;          // z - conj(w)
        const float imA = 1.0f / (dr * dr + diA * diA);
        const float imB = 1.0f / (dr * dr + diB * diB);
        const float e1r = dr * imA, e1i = -diA * imA;
        const float e2r = dr * imB, e2i = -diB * imB;
        // acc += v*e1 + conj(v)*e2
        const float fr = e1r + e2r, fi = e1i + e2i;
        const float gr = e1r - e2r, gi = e2i - e1i;
        float vr, vi;
        vr = sh[2][n]; vi = sh[3][n]; r00r += vr * fr + vi * gi; r00i += vr * fi + vi * gr;
        vr = sh[4][n]; vi = sh[5][n]; r01r += vr * fr + vi * gi; r01i += vr * fi + vi * gr;
        vr = sh[6][n]; vi = sh[7][n]; r10r += vr * fr + vi * gi; r10i += vr * fi + vi * gr;
        vr = sh[8][n]; vi = sh[9][n]; r11r += vr * fr + vi * gi; r11i += vr * fi + vi * gr;
    }
    r00r *= dt; r00i *= dt; r01r *= dt; r01i *= dt;
    r10r *= dt; r10i *= dt; r11r *= dt; r11i *= dt;

    // Woodbury: kf = r00 - r01*r10/(1 + r11)
    const float tr = r01r * r10r - r01i * r10i;
    const float ti = r01r * r10i + r01i * r10r;
    const float der = 1.0f + r11r, dei = r11i;
    const float dm = 1.0f / (der * der + dei * dei);
    const float kfr = r00r - (tr * der + ti * dei) * dm;
    const float kfi = r00i - (ti * der - tr * dei) * dm;

    // kf *= 2/(1+omega)
    const float fr2 = 2.0f * dr1 * invm, fi2 = -2.0f * di1 * invm;
    A[h * KP + 2 * l]     = kfr * fr2 - kfi * fi2;
    A[h * KP + 2 * l + 1] = kfr * fi2 + kfi * fr2;
}

// ---------------------------------------------------------------------------
// Stage 2: irfft as f32 GEMM with V_WMMA_F32_16X16X4_F32 + TDM-staged B panel.
// out[h,t] = sum_k A[h,k] * Bb[k,t];  M=256, N=2048, K=2052.
// One block per 16-wide N tile (128 blocks, 8 waves). The block's entire
// B panel (KP x 16 f32 = 131,328 B) is DMA'd to LDS once with
// tensor_load_to_lds (TENSORcnt), then each wave computes two 16x16 M tiles
// per k-step sharing one LDS B fragment (2 WMMA per B read).
// ---------------------------------------------------------------------------
__global__ void ssk_irfft_gemm(const float* __restrict__ A,
                               const float* __restrict__ Bb,
                               float* __restrict__ out) {
    extern __shared__ float ldsB[];          // [KP][16]
    const int tid  = threadIdx.x;
    const int lane = tid & 31;
    const int wave = tid >> 5;               // 0..7
    const int nt   = blockIdx.x;             // N tile (0..127)

    // --- TDM: DMA the KP x 16 panel Bb[:, nt*16 .. nt*16+15] into LDS ------
    if (tid < 32) {                          // one DMA per block (wave 0)
        const unsigned long long gaddr =
            (unsigned long long)(const void*)(Bb + (size_t)nt * 16);
        const unsigned int lds_off =
            (unsigned int)(unsigned long long)(const void*)ldsB; // low 32b = LDS offset
        u32x4 g0;
        g0[0] = 1u;                                   // count=1 (valid user D#)
        g0[1] = lds_off;                              // lds_addr
        g0[2] = (unsigned int)gaddr;                  // global_addr[31:0]
        g0[3] = (unsigned int)((gaddr >> 32) & 0x01FFFFFFull) | (2u << 30); // [56:32] | type=2
        i32x8 g1;
        g1[0] = 0x20000;                              // data_size=2 (4 bytes), no multicast
        g1[1] = (int)((LL & 0xFFFF) << 16);           // tensor_dim0 lo16 (2048)
        g1[2] = (int)((LL >> 16) | ((KP & 0xFFFF) << 16));  // t0 hi16 | tensor_dim1 lo16
        g1[3] = (int)((KP >> 16) | (16 << 16));       // t1 hi16 | tile_dim0=16
        g1[4] = (int)KP;                              // tile_dim1=2052, tile_dim2=0
        g1[5] = (int)LL;                              // tensor_dim0_stride lo32 (2048)
        g1[6] = 0;                                    // t0stride hi | t1stride lo
        g1[7] = 0;
        i32x4 z4 = {};
#if __has_include(<hip/amd_detail/amd_gfx1250_TDM.h>)
        i32x8 z8 = {};
        __builtin_amdgcn_tensor_load_to_lds(g0, g1, z4, z4, z8, 0);
#else
        __builtin_amdgcn_tensor_load_to_lds(g0, g1, z4, z4, 0);
#endif
        __builtin_amdgcn_s_wait_tensorcnt(0);
    }
    __syncthreads();                          // publish LDS panel to all 8 waves

    // --- WMMA: each wave computes M tiles mt0 = 2*wave, mt1 = 2*wave+1 -----
    const int row  = lane & 15;               // A row within tile / B column
    const int koff = (lane >> 4) << 1;        // 0 or 2 (16x4 f32 A/B lane layout)
    const float* Arow0 = A + ((2 * wave)     * 16 + row) * KP + koff;
    const float* Arow1 = A + ((2 * wave + 1) * 16 + row) * KP + koff;

    v8f acc0 = {};
    v8f acc1 = {};
    for (int k = 0; k < KP; k += 4) {
        const float2 a0v = *reinterpret_cast<const float2*>(Arow0 + k);
        const float2 a1v = *reinterpret_cast<const float2*>(Arow1 + k);
        v2f b;
        b.x = ldsB[(k + koff)     * 16 + row];
        b.y = ldsB[(k + koff + 1) * 16 + row];
        v2f a0; a0.x = a0v.x; a0.y = a0v.y;
        v2f a1; a1.x = a1v.x; a1.y = a1v.y;
        acc0 = __builtin_amdgcn_wmma_f32_16x16x4_f32(
            false, a0, false, b, (short)0, acc0, false, false);
        acc1 = __builtin_amdgcn_wmma_f32_16x16x4_f32(
            false, a1, false, b, (short)0, acc1, false, false);
    }

    // C/D layout: VGPR j -> M = j (lanes 0-15) or 8+j (lanes 16-31); N = lane&15
    const int col   = nt * 16 + row;
    const int mhalf = (lane >> 4) << 3;
    const int mb0 = (2 * wave)     * 16 + mhalf;
    const int mb1 = (2 * wave + 1) * 16 + mhalf;
    #pragma unroll
    for (int j = 0; j < 8; ++j) {
        out[(mb0 + j) * LL + col] = acc0[j];
        out[(mb1 + j) * LL + col] = acc1[j];
    }
}

extern "C" void kernel_launch(void* const* d_in, const int* in_sizes, int n_in,
                              void* d_out, int out_size, void* d_ws, size_t ws_size,
                              hipStream_t stream) {
    const float* Cm  = (const float*)d_in[0];  // (1,256,64,2)
    const float* Bm  = (const float*)d_in[1];  // (1,256,64,2)
    const float* Pm  = (const float*)d_in[2];  // (1,256,64,2)
    const float* iwr = (const float*)d_in[3];  // (256,64)
    const float* wim = (const float*)d_in[4];  // (256,64)
    const float* ldt = (const float*)d_in[5];  // (256,)
    // d_in[6] = L (int scalar) == 2048, compile-time constant here.

    float* Amat   = (float*)d_ws;                                   // 256*2052 f32
    float* Bbasis = (float*)((char*)d_ws + (size_t)HH * KP * 4);    // 2052*2048 f32
    float* out    = (float*)d_out;                                  // (1,256,2048)

    // Stage 0: DFT basis (independent of inputs)
    {
        const int total = KP * LL;
        ssk_build_basis<<<(total + 255) / 256, 256, 0, stream>>>(Bbasis);
    }
    // Stage 1: Cauchy + Woodbury spectrum -> Amat
    {
        dim3 grid(HH, (LH + 1 + 255) / 256);   // covers l in [0, 1025]
        ssk_cauchy<<<grid, 256, 0, stream>>>(Cm, Bm, Pm, iwr, wim, ldt, Amat);
    }
    // Stage 2: irfft via f32 WMMA GEMM with TDM-staged B panel -> out
    {
        const size_t lds_bytes = (size_t)KP * 16 * sizeof(float);   // 131,328 B
        (void)hipFuncSetAttribute((const void*)ssk_irfft_gemm,
                                  hipFuncAttributeMaxDynamicSharedMemorySize,
                                  (int)lds_bytes);
        ssk_irfft_gemm<<<LL / 16, 256, lds_bytes, stream>>>(Amat, Bbasis, out);
    }
}